// DynamicUpampling_65343632441724
// MI455X (gfx1250) — compile-verified
//
#include <hip/hip_runtime.h>
#include <stdint.h>

typedef __attribute__((ext_vector_type(2))) float v2f;
typedef __attribute__((ext_vector_type(4))) float v4f;
typedef __attribute__((ext_vector_type(8))) float v8f;

#define K_SIZE 3
#define SCALE  4
#define T_DIM  3
#define C_DIM  3
#define H_DIM  180
#define W_DIM  320
#define KTAPS  27                  // t*k*k taps
#define NROWS  (SCALE*SCALE*K_SIZE*K_SIZE*T_DIM)   // 432 (ch,t) rows / pixel
#define TILE_W 32
#define XS_W   (TILE_W + 2)
#define MROW   28                  // taps per m-row, padded (slot 27 = 0)
#define PIXSTRIDE (16*MROW + 4)    // 452 dwords: 16B-aligned, banks skewed by 4
#define PT_CH  (C_DIM + 1)         // extra all-zero channel for N>=3 lanes

__global__ __launch_bounds__(256)
void duf_wmma_kernel(const float* __restrict__ x,
                     const float* __restrict__ kern,
                     float* __restrict__ out)
{
    const int wt = blockIdx.x;
    const int h  = blockIdx.y;
    const int b  = blockIdx.z;
    const int w0 = wt * TILE_W;

    const int tid  = threadIdx.x;       // 256 = 8 wave32
    const int lane = tid & 31;
    const int wv   = tid >> 5;

    const int HW = H_DIM * W_DIM;

    __shared__ __align__(16) float ker_s[TILE_W * PIXSTRIDE];        // ~57.9 KB
    __shared__ float xs[C_DIM * T_DIM * K_SIZE * XS_W];              // ~3.7 KB
    __shared__ __align__(16) float pt[PT_CH * TILE_W * MROW];        // ~14.3 KB

    // ---------- phase 1a: kernel tensor -> LDS via async DMA copies ----------
    // global row = ch*3 + t, ch = m*9 + (k1*3+k2); LDS slot = [p][m*28 + t*9+rem]
    // per-lane scattered LDS dest + coalesced 128B global rows, ASYNCcnt-tracked.
    {
        const float* kp = kern + (size_t)(b * NROWS) * HW
                               + (size_t)h * W_DIM + w0 + lane;
        for (int row = wv; row < NROWS; row += 8) {       // row: wave-uniform (SALU)
            const int t   = row % 3;
            const int q   = row / 3;
            const int mm  = q / 9;
            const int rem = q % 9;
            const int k   = t * 9 + rem;
            uint32_t lds_off = (uint32_t)(uintptr_t)
                               &ker_s[lane * PIXSTRIDE + mm * MROW + k];
            uint64_t gaddr   = (uint64_t)(uintptr_t)(kp + (size_t)row * HW);
            asm volatile("global_load_async_to_lds_b32 %0, %1, off th:TH_LOAD_NT"
                         :: "v"(lds_off), "v"(gaddr) : "memory");
        }
    }
    // zero the k==27 pad slots (so mean-sum over 28 == sum over 27, A pad handled)
    for (int i = tid; i < TILE_W * 16; i += 256)
        ker_s[(i >> 4) * PIXSTRIDE + (i & 15) * MROW + KTAPS] = 0.f;

    // ---------- phase 1b: x halo tile (replicate-padded) ----------
    for (int i = tid; i < C_DIM * T_DIM * K_SIZE * XS_W; i += 256) {
        int wl = i % XS_W;
        int r  = (i / XS_W) % K_SIZE;
        int t  = (i / (XS_W * K_SIZE)) % T_DIM;
        int c  =  i / (XS_W * K_SIZE * T_DIM);
        int gh = h + r - 1;   gh = gh < 0 ? 0 : (gh >= H_DIM ? H_DIM - 1 : gh);
        int gw = w0 + wl - 1; gw = gw < 0 ? 0 : (gw >= W_DIM ? W_DIM - 1 : gw);
        xs[i] = x[(size_t)((b * C_DIM + c) * T_DIM + t) * HW
                  + (size_t)gh * W_DIM + gw];
    }
    // make this wave's async LDS writes complete before the block barrier
    asm volatile("s_wait_asynccnt 0" ::: "memory");
    __syncthreads();

    // ---------- phase 1c: expand halo -> pt[c][p][k] (tap-linear, + zero ch) ---
    for (int i = tid; i < PT_CH * TILE_W * MROW; i += 256) {
        int k = i % MROW;
        int p = (i / MROW) % TILE_W;
        int c =  i / (MROW * TILE_W);
        float v = 0.f;
        if (c < C_DIM && k < KTAPS) {
            int t = k / 9, rem = k % 9, k1 = rem / 3, k2 = rem % 3;
            v = xs[((c * T_DIM + t) * K_SIZE + k1) * XS_W + p + k2];
        }
        pt[i] = v;
    }
    __syncthreads();

    // ---------- phase 2: per-pixel D(16x16) = A(16x28) x B(28x16) ----------
    const int m    = lane & 15;              // A row (subpixel) / B col (channel)
    const int half = lane >> 4;              // f32 frag: K = 4*s + 2*half + q
    const int ceff = (m < C_DIM) ? m : C_DIM;

    for (int pi = 0; pi < 4; ++pi) {
        const int p = wv * 4 + pi;

        const v4f* arow = reinterpret_cast<const v4f*>(&ker_s[p * PIXSTRIDE + m * MROW]);
        const v4f* brow = reinterpret_cast<const v4f*>(&pt[(ceff * TILE_W + p) * MROW]);
        v4f ar[7], br[7];
        #pragma unroll
        for (int s = 0; s < 7; ++s) { ar[s] = arow[s]; br[s] = brow[s]; }

        float sum = 0.f;
        #pragma unroll
        for (int s = 0; s < 7; ++s) sum += (ar[s].x + ar[s].y) + (ar[s].z + ar[s].w);
        const float c0 = (1.0f - sum) * (1.0f / KTAPS);   // -mean + 1/K

        v8f acc = {};
        #pragma unroll
        for (int s = 0; s < 7; ++s) {
            v2f a, bf;
            a.x = (half ? ar[s].z : ar[s].x) + c0;
            if (s == 6)  // k = 27 pad lives in hi-half .w of the last step
                a.y = half ? 0.f : (ar[s].y + c0);
            else
                a.y = (half ? ar[s].w : ar[s].y) + c0;
            bf.x = half ? br[s].z : br[s].x;
            bf.y = half ? br[s].w : br[s].y;     // br slot 27 is a true zero
            acc = __builtin_amdgcn_wmma_f32_16x16x4_f32(
                      false, a, false, bf, (short)0, acc, false, false);
        }

        // acc[j] = D[j + 8*half, m]: j=0..3 -> (s1=2*half,   s2=0..3)
        //                            j=4..7 -> (s1=2*half+1, s2=0..3)
        if (m < C_DIM) {
            const int w  = w0 + p;
            const int OH = H_DIM * SCALE, OW = W_DIM * SCALE;
            float* o0 = out + (size_t)((b * C_DIM + m) * OH + h * SCALE + 2 * half) * OW
                            + (size_t)w * SCALE;
            v4f r0 = { acc[0], acc[1], acc[2], acc[3] };
            v4f r1 = { acc[4], acc[5], acc[6], acc[7] };
            __builtin_nontemporal_store(r0, (v4f*)o0);
            __builtin_nontemporal_store(r1, (v4f*)(o0 + OW));
        }
    }
}

extern "C" void kernel_launch(void* const* d_in, const int* in_sizes, int n_in,
                              void* d_out, int out_size, void* d_ws, size_t ws_size,
                              hipStream_t stream) {
    (void)n_in; (void)out_size; (void)d_ws; (void)ws_size;
    const float* x    = (const float*)d_in[0];
    const float* kern = (const float*)d_in[1];
    float* out        = (float*)d_out;

    const int B = in_sizes[0] / (C_DIM * T_DIM * H_DIM * W_DIM);  // = 4
    dim3 grid(W_DIM / TILE_W, H_DIM, B);
    duf_wmma_kernel<<<grid, 256, 0, stream>>>(x, kern, out);
}